// Denoise_Net_52759378264425
// MI455X (gfx1250) — compile-verified
//
#include <hip/hip_runtime.h>
#include <hip/hip_bf16.h>
#include <math.h>

#define FD 128

typedef float v2f __attribute__((ext_vector_type(2)));
typedef float v4f __attribute__((ext_vector_type(4)));
typedef float v8f __attribute__((ext_vector_type(8)));

// ---------------------------------------------------------------------------
// GEMM: out[i][j] = sum_k X[i][k] * W[j][k] + (bias ? bias[j] : 0)
// X: [nRows,128], W: [128,128] row-major, out: [nRows,128]
// 256 threads = 8 waves; wave w computes columns [16w,16w+16).
// Uses V_WMMA_F32_16X16X4_F32 (fp32 matrix core), 32 steps over K=128.
// ---------------------------------------------------------------------------
__global__ __launch_bounds__(256) void gemm_xwt(
    const float* __restrict__ X, const float* __restrict__ W,
    const float* __restrict__ bias, float* __restrict__ out, int nRows)
{
    __shared__ float Xs[16][FD + 4];     // pad 4 -> stride 132 (conflict-free)
    const int lane = threadIdx.x & 31;
    const int wave = threadIdx.x >> 5;
    const int lh   = lane & 15;          // lane within half
    const int hi   = lane >> 4;          // 0: K+0..1 | 1: K+2..3 (ISA A/B layout)
    const int j0   = wave * 16;

    // Preload B fragments for this wave's 16 columns of W (kept in VGPRs).
    // B[k][n] = W[j0+n][k]; VGPR0 holds K=base+2*hi, VGPR1 holds K=base+2*hi+1.
    v2f bfrag[32];
    {
        const float* wrow = W + (size_t)(j0 + lh) * FD;
        #pragma unroll
        for (int kk = 0; kk < 32; ++kk)
            bfrag[kk] = *(const v2f*)(wrow + kk * 4 + hi * 2);
    }

    const int numTiles = (nRows + 15) >> 4;
    for (int tile = blockIdx.x; tile < numTiles; tile += gridDim.x) {
        const int m0 = tile << 4;
        __syncthreads();
        // Stage 16x128 tile of X into LDS (float4, coalesced).
        for (int i = threadIdx.x; i < 16 * 32; i += 256) {
            const int r = i >> 5, c4 = (i & 31) << 2;
            const int rr = m0 + r;
            v4f v = {0.f, 0.f, 0.f, 0.f};
            if (rr < nRows) v = *(const v4f*)(X + (size_t)rr * FD + c4);
            *(v4f*)&Xs[r][c4] = v;
        }
        __syncthreads();

        v8f acc = {0.f, 0.f, 0.f, 0.f, 0.f, 0.f, 0.f, 0.f};
        #pragma unroll
        for (int kk = 0; kk < 32; ++kk) {
            // A[16x4]: lane L<16 -> row L, K=base..base+1; L>=16 -> row L-16, K=base+2..3
            v2f a = *(const v2f*)(&Xs[lh][kk * 4 + hi * 2]);
            acc = __builtin_amdgcn_wmma_f32_16x16x4_f32(
                false, a, false, bfrag[kk], (short)0, acc, false, false);
        }

        // D layout: VGPR v -> (M=v, N=lane) for lanes 0-15; (M=v+8, N=lane-16) for 16-31
        const int col = j0 + lh;
        const float bv = bias ? bias[col] : 0.0f;
        const int rbase = m0 + hi * 8;
        #pragma unroll
        for (int v = 0; v < 8; ++v) {
            const int rr = rbase + v;
            if (rr < nRows) out[(size_t)rr * FD + col] = acc[v] + bv;
        }
    }
}

// ---------------------------------------------------------------------------
// Degree / normalization kernels. deg[i] = 1 (self loop) + #{e: col[e]==i};
// then dinv[i] = 1/sqrt(deg[i])  (deg >= 1 always).
// ---------------------------------------------------------------------------
__global__ void k_deg_init(float* deg, int n) {
    int i = blockIdx.x * blockDim.x + threadIdx.x;
    if (i < n) deg[i] = 1.0f;
}
__global__ void k_deg_count(const long long* __restrict__ col, float* deg, int nE) {
    int i = blockIdx.x * blockDim.x + threadIdx.x;
    if (i < nE) atomicAdd(&deg[(int)col[i]], 1.0f);
}
__global__ void k_deg_fin(float* deg, int n) {
    int i = blockIdx.x * blockDim.x + threadIdx.x;
    if (i < n) deg[i] = 1.0f / sqrtf(deg[i]);
}

// ---------------------------------------------------------------------------
// Aggregation init: out[i][:] = X[i][:] * dinv[i]^2 (self-loop term) + bias
// (bias is the post-aggregation GCN bias)
// ---------------------------------------------------------------------------
__global__ void k_agg_init(const float* __restrict__ X, const float* __restrict__ dinv,
                           const float* __restrict__ bias, float* __restrict__ out, int n) {
    int idx = blockIdx.x * blockDim.x + threadIdx.x;
    if (idx >= n * FD) return;
    const int i = idx >> 7, d = idx & 127;
    const float di = dinv[i];
    out[idx] = X[idx] * di * di + bias[d];
}

// ---------------------------------------------------------------------------
// Edge scatter: one wave32 per edge; each lane moves a float4 (32*4 = 128).
// out[col] += X[row] * dinv[row]*dinv[col]  via global_atomic_add_f32.
// ---------------------------------------------------------------------------
__global__ __launch_bounds__(256) void k_scatter(
    const float* __restrict__ X, const float* __restrict__ dinv,
    const long long* __restrict__ rowI, const long long* __restrict__ colI,
    float* __restrict__ out, int nE)
{
    const int gid = blockIdx.x * blockDim.x + threadIdx.x;
    const int e = gid >> 5;
    const int lane = gid & 31;
    if (e >= nE) return;
    const int r = (int)rowI[e];
    const int c = (int)colI[e];
    const float w = dinv[r] * dinv[c];
    const v4f v = *(const v4f*)(X + (size_t)r * FD + lane * 4);
    float* dst = out + (size_t)c * FD + lane * 4;
    atomicAdd(dst + 0, v.x * w);
    atomicAdd(dst + 1, v.y * w);
    atomicAdd(dst + 2, v.z * w);
    atomicAdd(dst + 3, v.w * w);
}

// ---------------------------------------------------------------------------
// Time-MLP stage 1: g[i][d] = gelu_exact(t[i]*Wt1[d] + bt1[d])
// ---------------------------------------------------------------------------
__global__ void k_tgelu(const int* __restrict__ t, const float* __restrict__ Wt1,
                        const float* __restrict__ bt1, float* __restrict__ g, int n) {
    int idx = blockIdx.x * blockDim.x + threadIdx.x;
    if (idx >= n * FD) return;
    const int i = idx >> 7, d = idx & 127;
    const float x = (float)t[i] * Wt1[d] + bt1[d];
    g[idx] = 0.5f * x * (1.0f + erff(x * 0.70710678118654752f));
}

// ---------------------------------------------------------------------------
// h1 = elu(agg1 + t_emb), in place on agg
// ---------------------------------------------------------------------------
__global__ void k_add_elu(float* __restrict__ agg, const float* __restrict__ temb, int total) {
    int idx = blockIdx.x * blockDim.x + threadIdx.x;
    if (idx >= total) return;
    const float x = agg[idx] + temb[idx];
    agg[idx] = (x > 0.0f) ? x : expm1f(x);
}

// ---------------------------------------------------------------------------
extern "C" void kernel_launch(void* const* d_in, const int* in_sizes, int n_in,
                              void* d_out, int out_size, void* d_ws, size_t ws_size,
                              hipStream_t stream) {
    const float*     z   = (const float*)d_in[0];
    const long long* ei  = (const long long*)d_in[1];
    const int*       t   = (const int*)d_in[2];
    const float*     Wt1 = (const float*)d_in[3];
    const float*     bt1 = (const float*)d_in[4];
    const float*     Wt2 = (const float*)d_in[5];
    const float*     bt2 = (const float*)d_in[6];
    const float*     W1  = (const float*)d_in[7];
    const float*     b1  = (const float*)d_in[8];
    const float*     W2  = (const float*)d_in[9];
    const float*     b2  = (const float*)d_in[10];
    float* out = (float*)d_out;

    const int n  = in_sizes[0] / FD;       // 50000
    const int nE = in_sizes[1] / 2;        // 640000
    const long long* rowI = ei;            // edge_index[0]
    const long long* colI = ei + nE;       // edge_index[1]

    // workspace layout
    char* ws = (char*)d_ws;
    const size_t AL = 256;
    const size_t ndBytes = (((size_t)n * FD * sizeof(float)) + AL - 1) & ~(AL - 1);
    float* dinv = (float*)ws;
    size_t off = (((size_t)n * sizeof(float)) + AL - 1) & ~(AL - 1);
    float* bufA = (float*)(ws + off); off += ndBytes;
    float* bufB = (float*)(ws + off); off += ndBytes;
    float* bufC = (float*)(ws + off);

    const int total = n * FD;
    dim3 blk(256);
    const int gElem  = (total + 255) / 256;
    const int gNode  = (n + 255) / 256;
    const int gEdge  = (nE + 255) / 256;
    const int gScat  = (nE * 32 + 255) / 256;   // one wave per edge
    const int gGemm  = 640;

    // symmetric normalization coefficients
    k_deg_init <<<gNode, blk, 0, stream>>>(dinv, n);
    k_deg_count<<<gEdge, blk, 0, stream>>>(colI, dinv, nE);
    k_deg_fin  <<<gNode, blk, 0, stream>>>(dinv, n);

    // conv1 linear: bufA = z @ W1^T
    gemm_xwt<<<gGemm, blk, 0, stream>>>(z, W1, nullptr, bufA, n);
    // conv1 aggregate into bufB (self loop + bias b1, then edge scatter)
    k_agg_init<<<gElem, blk, 0, stream>>>(bufA, dinv, b1, bufB, n);
    k_scatter <<<gScat, blk, 0, stream>>>(bufA, dinv, rowI, colI, bufB, nE);

    // time MLP: bufC = gelu(t*Wt1+bt1); bufA = bufC @ Wt2^T + bt2
    k_tgelu <<<gElem, blk, 0, stream>>>(t, Wt1, bt1, bufC, n);
    gemm_xwt<<<gGemm, blk, 0, stream>>>(bufC, Wt2, bt2, bufA, n);

    // h1 = elu(agg1 + t_emb), in place in bufB
    k_add_elu<<<gElem, blk, 0, stream>>>(bufB, bufA, total);

    // conv2 linear: bufC = h1 @ W2^T ; aggregate directly into d_out (+b2)
    gemm_xwt  <<<gGemm, blk, 0, stream>>>(bufB, W2, nullptr, bufC, n);
    k_agg_init<<<gElem, blk, 0, stream>>>(bufC, dinv, b2, out, n);
    k_scatter <<<gScat, blk, 0, stream>>>(bufC, dinv, rowI, colI, out, nE);
}